// GumbelVectorQuantizer_11802570130110
// MI455X (gfx1250) — compile-verified
//
#include <hip/hip_runtime.h>
#include <math.h>

typedef float v2f __attribute__((ext_vector_type(2)));
typedef float v8f __attribute__((ext_vector_type(8)));

#define NTOTAL 24576        // B*T
#define DIMK   512
#define GV     640          // G*V
#define VV     320          // V per group
#define VARD   128
#define LPITCH 644          // padded logits pitch (bank-conflict-free stores)
#define OUT_ELEMS (NTOTAL * 256)

__device__ __forceinline__ float wave_sum(float v) {
#pragma unroll
  for (int o = 16; o; o >>= 1) v += __shfl_xor(v, o, 32);
  return v;
}

__device__ __forceinline__ void wave_argmax(float& val, int& idx) {
#pragma unroll
  for (int o = 16; o; o >>= 1) {
    float ov = __shfl_xor(val, o, 32);
    int   oi = __shfl_xor(idx, o, 32);
    if (ov > val || (ov == val && oi < idx)) { val = ov; idx = oi; }
  }
}

// One block = 16 tokens. 8 waves: GEMM (WMMA f32 16x16x4, 5 independent
// accumulator chains per wave, shared A fragment) -> LDS logits tile,
// then per-(token,group) softmax/argmax/gather, block-level [G,V] partials.
__global__ __launch_bounds__(256) void gvq_main(
    const float* __restrict__ x, const float* __restrict__ Wm,
    const float* __restrict__ bias, const float* __restrict__ codebook,
    const float* __restrict__ gumbel, float* __restrict__ out,
    float* __restrict__ g_soft, float* __restrict__ g_hard) {
  __shared__ float s_logits[16 * LPITCH];
  __shared__ float s_soft[GV];
  __shared__ float s_hard[GV];

  const int tid  = threadIdx.x;
  const int wave = tid >> 5, lane = tid & 31;
  const int mrow = lane & 15, khalf = lane >> 4;
  const int t0   = blockIdx.x * 16;

  for (int i = tid; i < GV; i += 256) { s_soft[i] = 0.f; s_hard[i] = 0.f; }

  // ---- GEMM: logits[16 x 640] = x_tile @ W^T via V_WMMA_F32_16X16X4_F32 ----
  // A 16x4 frag: lane<16 -> M=lane, K=k..k+1 ; lane>=16 -> M=lane-16, K=k+2..k+3
  // One wave owns 5 column tiles (80 outputs); A frag shared across tiles,
  // 5 independent WMMA accumulator chains (no back-to-back D->C hazard).
  const float* xrow  = x + (size_t)(t0 + mrow) * DIMK + khalf * 2;
  const int    cbase = wave * 80;                 // first output column
  const float* wbase = Wm + (size_t)(cbase + mrow) * DIMK + khalf * 2;

  v8f acc[5];
#pragma unroll
  for (int t = 0; t < 5; ++t) acc[t] = (v8f){0.f,0.f,0.f,0.f,0.f,0.f,0.f,0.f};

#pragma unroll 2
  for (int k = 0; k < DIMK; k += 4) {
    const v2f afrag = *(const v2f*)(xrow + k);
#pragma unroll
    for (int t = 0; t < 5; ++t) {
      // tile t is a constant +t*32KB immediate offset (fits 24-bit IOFFSET)
      const v2f bfrag = *(const v2f*)(wbase + k + t * 16 * DIMK);
      acc[t] = __builtin_amdgcn_wmma_f32_16x16x4_f32(
          false, afrag, false, bfrag, (short)0, acc[t], false, false);
    }
  }

#pragma unroll
  for (int t = 0; t < 5; ++t) {
    const int c0 = cbase + t * 16;
    const float bv = bias[c0 + mrow];
#pragma unroll
    for (int r = 0; r < 8; ++r)   // D: VGPR r -> row r (+8 for lanes 16..31)
      s_logits[(r + khalf * 8) * LPITCH + c0 + mrow] = acc[t][r] + bv;
  }
  __syncthreads();

  // ---- Per (token, group): 32 pairs, 4 per wave ----
#pragma unroll 1
  for (int i = 0; i < 4; ++i) {
    const int p = wave * 4 + i;
    const int tok = p >> 1, g = p & 1;
    const float* lrow = &s_logits[tok * LPITCH + g * VV];
    const size_t ng = (size_t)(t0 + tok) * 2 + g;
    const float* grow = gumbel + ng * VV;

    float lv[10], nv[10];
    float mraw = -3.4e38f, mnoi = -3.4e38f;
    int araw = lane, anoi = lane;
#pragma unroll
    for (int j = 0; j < 10; ++j) {
      const int v = lane + j * 32;
      const float l = lrow[v];
      const float n = l + grow[v];
      lv[j] = l; nv[j] = n;
      if (l > mraw) { mraw = l; araw = v; }
      if (n > mnoi) { mnoi = n; anoi = v; }
    }
    wave_argmax(mraw, araw);   // argmax of raw logits (code perplexity)
    wave_argmax(mnoi, anoi);   // argmax of logits+gumbel (selection)

    float er[10], sraw = 0.f, snoi = 0.f;
#pragma unroll
    for (int j = 0; j < 10; ++j) {
      er[j] = __expf(lv[j] - mraw);
      sraw += er[j];
      snoi += __expf((nv[j] - mnoi) * 0.5f);   // 1/TAU = 0.5
    }
    sraw = wave_sum(sraw);
    snoi = wave_sum(snoi);

    const float inv_sraw = 1.f / sraw;
#pragma unroll
    for (int j = 0; j < 10; ++j)               // avg_probs partials
      atomicAdd(&s_soft[g * VV + lane + j * 32], er[j] * inv_sraw);
    if (lane == 0) atomicAdd(&s_hard[g * VV + araw], 1.f);

    // y value at selected index: (1 + s) - s in fp32; elsewhere exactly 0.
    const float sidx = 1.f / snoi;             // softmax value at its argmax
    const float ysel = (1.f + sidx) - sidx;
    const float4* cbr = (const float4*)(codebook + ((size_t)g * VV + anoi) * VARD);
    float4* orow = (float4*)(out + ng * VARD);
    float4 cv = cbr[lane];
    cv.x *= ysel; cv.y *= ysel; cv.z *= ysel; cv.w *= ysel;
    orow[lane] = cv;
  }
  __syncthreads();

  for (int i = tid; i < GV; i += 256) {
    atomicAdd(&g_soft[i], s_soft[i]);
    atomicAdd(&g_hard[i], s_hard[i]);
  }
}

__global__ __launch_bounds__(64) void gvq_finalize(
    const float* __restrict__ g_soft, const float* __restrict__ g_hard,
    float* __restrict__ scalars) {
  __shared__ float part[4];
  const int wave = threadIdx.x >> 5, lane = threadIdx.x & 31;  // wave == group
  const float invN = 1.f / (float)NTOTAL;
  float eh = 0.f, es = 0.f;
  for (int v = lane; v < VV; v += 32) {
    const float ph = g_hard[wave * VV + v] * invN;
    const float ps = g_soft[wave * VV + v] * invN;
    eh += ph * logf(ph + 1e-7f);
    es += ps * logf(ps + 1e-7f);
  }
  eh = wave_sum(eh);
  es = wave_sum(es);
  if (lane == 0) { part[wave * 2] = eh; part[wave * 2 + 1] = es; }
  __syncthreads();
  if (threadIdx.x == 0) {
    scalars[0] = expf(-part[0]) + expf(-part[2]);  // code_perplexity
    scalars[1] = expf(-part[1]) + expf(-part[3]);  // prob_perplexity
  }
}

extern "C" void kernel_launch(void* const* d_in, const int* in_sizes, int n_in,
                              void* d_out, int out_size, void* d_ws, size_t ws_size,
                              hipStream_t stream) {
  (void)in_sizes; (void)n_in; (void)out_size; (void)ws_size;
  const float* x    = (const float*)d_in[0];  // [24576, 512]
  const float* Wm   = (const float*)d_in[1];  // [640, 512]
  const float* bias = (const float*)d_in[2];  // [640] (zeros)
  const float* cbk  = (const float*)d_in[3];  // [640, 128]
  const float* gum  = (const float*)d_in[4];  // [24576, 2, 320]
  float* out = (float*)d_out;

  float* ws_soft = (float*)d_ws;        // [640] avg-prob sums
  float* ws_hard = ws_soft + GV;        // [640] hard counts

  hipMemsetAsync(d_ws, 0, 2 * GV * sizeof(float), stream);
  gvq_main<<<NTOTAL / 16, 256, 0, stream>>>(x, Wm, bias, cbk, gum, out,
                                            ws_soft, ws_hard);
  gvq_finalize<<<1, 64, 0, stream>>>(ws_soft, ws_hard, out + OUT_ELEMS);
}